// GetElements_38293928411433
// MI455X (gfx1250) — compile-verified
//
#include <hip/hip_runtime.h>
#include <math.h>

// ---------------------------------------------------------------------------
// Point-NMS post-processing (GetElements) for gfx1250 / MI455X.
//   d2(i,j) = |pi|^2 + |pj|^2 - 2*pi.pj ; Gram tiles via V_WMMA_F32_16X16X4_F32
// ---------------------------------------------------------------------------

#define NQ          4096
#define NCLS_TILES  256          // 4096 / 16
#define MASK_W32    128          // 4096 bits / 32 per row
#define MASK_W16    256          // 4096 bits / 16 per row
#define TJ_CHUNK    32           // col tiles per wave strip
#define TJ_GRID     (NCLS_TILES / TJ_CHUNK)   // 8
#define THRESH2     25.0f        // 5.0^2
#define PT_SCORE    0.05f

typedef __attribute__((ext_vector_type(2))) float v2f;
typedef __attribute__((ext_vector_type(8))) float v8f;

// ---- Kernel 1: softmax + pixel scaling -----------------------------------
__global__ __launch_bounds__(256) void prep_kernel(
    const float* __restrict__ logits,   // [NQ,3]
    const float* __restrict__ boxes,    // [NQ,2]
    const float* __restrict__ tsz,      // [2] (w,h)
    float* __restrict__ out,            // [NQ,5]
    float* __restrict__ endS, float* __restrict__ junS,
    float* __restrict__ px,  float* __restrict__ py)
{
    int i = blockIdx.x * blockDim.x + threadIdx.x;
    if (i >= NQ) return;
    float l0 = logits[i * 3 + 0], l1 = logits[i * 3 + 1], l2 = logits[i * 3 + 2];
    float m  = fmaxf(l0, fmaxf(l1, l2));
    float e0 = __expf(l0 - m), e1 = __expf(l1 - m), e2 = __expf(l2 - m);
    float inv = 1.0f / (e0 + e1 + e2);
    float p0 = e0 * inv, p1 = e1 * inv, p2 = e2 * inv;
    float x = boxes[i * 2 + 0] * tsz[0];
    float y = boxes[i * 2 + 1] * tsz[1];
    out[i * 5 + 0] = 1.0f - p2;   // tgt score
    out[i * 5 + 1] = x;
    out[i * 5 + 2] = y;
    endS[i] = p0;  junS[i] = p1;  px[i] = x;  py[i] = y;
}

// ---- Kernel 2: stable bitonic sort (descending) per class ----------------
__global__ __launch_bounds__(1024) void sort_kernel(
    const float* __restrict__ endS, const float* __restrict__ junS,
    const float* __restrict__ px,   const float* __restrict__ py,
    float* __restrict__ sortedBase) // per class: [ss NQ][sx NQ][sy NQ][sn NQ]
{
    const int cls = blockIdx.x;
    const float* S = cls ? junS : endS;
    __shared__ float key[NQ];
    __shared__ int   idx[NQ];
    const int tid = threadIdx.x;

    for (int e = tid; e < NQ; e += 1024) {
        float s = S[e];
        key[e] = (s >= PT_SCORE) ? s : -__builtin_inff();
        idx[e] = e;
    }
    __syncthreads();

    for (int k = 2; k <= NQ; k <<= 1) {
        for (int j = k >> 1; j > 0; j >>= 1) {
            for (int e = tid; e < NQ; e += 1024) {
                int ixj = e ^ j;
                if (ixj > e) {
                    float a = key[e], b = key[ixj];
                    int   ia = idx[e], ib = idx[ixj];
                    // descending, stable: does a rank before b?
                    bool aFirst    = (a > b) || ((a == b) && (ia < ib));
                    bool descBlock = ((e & k) == 0);
                    if (aFirst != descBlock) {
                        key[e] = b; key[ixj] = a;
                        idx[e] = ib; idx[ixj] = ia;
                    }
                }
            }
            __syncthreads();
        }
    }

    float* ss = sortedBase + (size_t)cls * 4 * NQ;
    float* sx = ss + NQ;
    float* sy = sx + NQ;
    float* sn = sy + NQ;
    for (int e = tid; e < NQ; e += 1024) {
        float s  = key[e];
        int   id = idx[e];
        float x = px[id], y = py[id];
        ss[e] = s; sx[e] = x; sy[e] = y; sn[e] = x * x + y * y;
    }
}

// ---- Kernel 3: suppression bitmask via WMMA Gram tiles (strip-mined) -----
// One wave per (row-tile, tj-chunk): A-matrix + row norms loaded once,
// loop over TJ_CHUNK column tiles reusing A. K-dim padding (2 -> 4) is done
// branchlessly: all lanes load, upper-half lanes multiply by 0 (no per-iter
// exec save/branch). Ballot results are wave-uniform; lanes 0..15 each store
// one row's 16-bit mask word in parallel.
__global__ __launch_bounds__(32) void nms_mask_kernel(
    const float* __restrict__ sortedBase,
    unsigned short* __restrict__ maskBase)
{
    const int cls = blockIdx.z;
    const float* sx = sortedBase + (size_t)cls * 4 * NQ + NQ;
    const float* sy = sx + NQ;
    const float* sn = sy + NQ;
    unsigned short* mask = maskBase + (size_t)cls * NQ * MASK_W16;

    const int ti   = blockIdx.y;          // row tile
    const int tj0  = blockIdx.x * TJ_CHUNK;
    const int lane = threadIdx.x;         // wave32
    const int half = lane >> 4;           // 0: K=0..1 live, 1: K=2..3 (zero pad)
    const int l16  = lane & 15;
    const int rowbase = ti << 4;
    const float kpad = (half == 0) ? 1.0f : 0.0f;   // branchless K padding

    // A-matrix (16x4 f32): lanes 0-15 hold K=0,1; lanes 16-31 are K=2,3 = 0.
    v2f a;
    a[0] = sx[rowbase + l16] * kpad;
    a[1] = sy[rowbase + l16] * kpad;

    // Per-lane row norms for the 8 C-VGPRs: VGPR k holds rows {k, k+8}.
    float nr[8];
#pragma unroll
    for (int k = 0; k < 8; ++k)
        nr[k] = sn[rowbase + k + (half << 3)];

    for (int t = 0; t < TJ_CHUNK; ++t) {
        const int tj      = tj0 + t;
        const int colbase = tj << 4;
        const int col     = colbase + l16;    // N = lane%16 in both halves

        v2f b;
        b[0] = sx[col] * kpad;                // B[K=0..1][N]; K=2,3 zeroed
        b[1] = sy[col] * kpad;

        v8f c = {};
        c = __builtin_amdgcn_wmma_f32_16x16x4_f32(
                /*neg_a=*/false, a, /*neg_b=*/false, b,
                /*c_mod=*/(short)0, c, /*reuse_a=*/false, /*reuse_b=*/false);

        const float ncol = sn[col];
        unsigned balv[8];
#pragma unroll
        for (int k = 0; k < 8; ++k) {
            const int   row  = rowbase + k + (half << 3);
            const float d2   = nr[k] + ncol - 2.0f * c[k];
            const bool  pred = (d2 < THRESH2) && (col > row);
            // bits[15:0] = row k cols, bits[31:16] = row k+8 cols
            balv[k] = __builtin_amdgcn_ballot_w32(pred);
        }
        // Distribute: lane L (0..15) stores row (rowbase+L):
        //   L<8 -> low half of balv[L]; L>=8 -> high half of balv[L-8].
        unsigned sel = balv[0];
#pragma unroll
        for (int k = 1; k < 8; ++k)
            if ((lane & 7) == k) sel = balv[k];
        const unsigned short h = (lane & 8) ? (unsigned short)(sel >> 16)
                                            : (unsigned short)(sel & 0xFFFFu);
        if (lane < 16)
            mask[(size_t)(rowbase + lane) * MASK_W16 + tj] = h;
    }
}

// ---- Kernel 4: greedy sequential scan + emit NMS columns -----------------
// Single wave per class: 32 lanes x 4 keep words; barriers in a one-wave
// workgroup lower to S_NOP (ISA 3.1) but keep compiler-visible ordering.
__global__ __launch_bounds__(32) void nms_scan_kernel(
    const float* __restrict__ sortedBase,
    const unsigned* __restrict__ maskBase,
    float* __restrict__ out)
{
    const int cls = blockIdx.x;
    const float*    ss   = sortedBase + (size_t)cls * 4 * NQ;
    const unsigned* mask = maskBase + (size_t)cls * NQ * MASK_W32;

    __shared__ unsigned keep[MASK_W32];
    const int tid = threadIdx.x;          // 0..31, owns words 4*tid .. 4*tid+3

#pragma unroll
    for (int w = 0; w < 4; ++w) {
        unsigned kw = 0;
        const int base = (tid * 4 + w) * 32;
        for (int b = 0; b < 32; ++b)
            if (ss[base + b] >= PT_SCORE) kw |= (1u << b);
        keep[tid * 4 + w] = kw;
    }
    __syncthreads();

    for (int i = 0; i < NQ; ++i) {
        const unsigned ki = (keep[i >> 5] >> (i & 31)) & 1u;
        __syncthreads();
        if (ki) {
            const uint4 m = ((const uint4*)(mask + (size_t)i * MASK_W32))[tid];
            keep[tid * 4 + 0] &= ~m.x;
            keep[tid * 4 + 1] &= ~m.y;
            keep[tid * 4 + 2] &= ~m.z;
            keep[tid * 4 + 3] &= ~m.w;
        }
        __syncthreads();
    }

    for (int e = tid; e < NQ; e += 32) {
        const bool kept = (keep[e >> 5] >> (e & 31)) & 1u;
        out[(size_t)e * 5 + 3 + cls] = kept ? ss[e] : 0.0f;
    }
}

// ---------------------------------------------------------------------------
extern "C" void kernel_launch(void* const* d_in, const int* in_sizes, int n_in,
                              void* d_out, int out_size, void* d_ws, size_t ws_size,
                              hipStream_t stream)
{
    const float* logits = (const float*)d_in[0];   // [1,4096,3]
    const float* boxes  = (const float*)d_in[1];   // [1,4096,2]
    // d_in[2] = pred_gids, unused by the reference output
    const float* tsz    = (const float*)d_in[3];   // [1,2]
    float* out = (float*)d_out;                    // [1,4096,5]

    // ---- workspace layout (floats, then masks) ----
    float* ws   = (float*)d_ws;
    float* endS = ws;                 // NQ
    float* junS = ws + NQ;            // NQ
    float* px   = ws + 2 * NQ;        // NQ
    float* py   = ws + 3 * NQ;        // NQ
    float* sortedBase = ws + 4 * NQ;  // 2 classes * 4*NQ floats
    unsigned* maskBase = (unsigned*)(ws + 12 * NQ); // 2 classes * NQ*MASK_W32 u32 (2 MB each)

    prep_kernel<<<(NQ + 255) / 256, 256, 0, stream>>>(
        logits, boxes, tsz, out, endS, junS, px, py);

    sort_kernel<<<2, 1024, 0, stream>>>(endS, junS, px, py, sortedBase);

    dim3 mgrid(TJ_GRID, NCLS_TILES, 2);
    nms_mask_kernel<<<mgrid, 32, 0, stream>>>(
        sortedBase, (unsigned short*)maskBase);

    nms_scan_kernel<<<2, 32, 0, stream>>>(sortedBase, maskBase, out);
}